// HATS_46626164966029
// MI455X (gfx1250) — compile-verified
//
#include <hip/hip_runtime.h>
#include <cstdint>
#include <cstddef>

// ---------------- HATS constants (match reference) ----------------
#define GRID_W   24          // GW
#define N_CELLS  432         // GH*GW
#define CELL_K   10          // K
#define RADIUS   3           // R
#define SURF     7           // 2R+1
#define TAU_INV  (1.0e-6f)   // 1/TAU
#define DELTA_TW (1.0e5f)    // DELTA_T
#define TMAX     2048        // max events per sequence staged in LDS
#define SLICES   8           // i-range slices per batch (parallelism)
#define NPART    8           // partial histogram copies (== SLICES)

// ---------- CDNA5 async global->LDS path (guarded; verified present) ----------
#if defined(__HIP_DEVICE_COMPILE__) && defined(__gfx1250__) && \
    __has_builtin(__builtin_amdgcn_global_load_async_to_lds_b128)
#define HATS_ASYNC 1
#else
#define HATS_ASYNC 0
#endif

#if HATS_ASYNC
typedef int hats_v4i __attribute__((ext_vector_type(4)));
typedef __attribute__((address_space(1))) hats_v4i* hats_gptr;
typedef __attribute__((address_space(3))) hats_v4i* hats_lptr;

__device__ __forceinline__ void hats_async_ld_b128(const float* g, float* l) {
  // emits global_load_async_to_lds_b128 (tracked by ASYNCcnt)
  __builtin_amdgcn_global_load_async_to_lds_b128(
      (hats_gptr)(g), (hats_lptr)(l), /*imm offset*/ 0, /*cpol*/ 0);
}
__device__ __forceinline__ void hats_wait_async0() {
#if __has_builtin(__builtin_amdgcn_s_wait_asynccnt)
  __builtin_amdgcn_s_wait_asynccnt(0);
#else
  asm volatile("s_wait_asynccnt 0" ::: "memory");
#endif
}
#endif

// ---------- CDNA5 WMMA path for the partial-histogram tree reduction ----------
#if defined(__HIP_DEVICE_COMPILE__) && defined(__gfx1250__) && \
    __has_builtin(__builtin_amdgcn_wmma_f32_16x16x4_f32)
#define HATS_WMMA 1
#else
#define HATS_WMMA 0
#endif

typedef float hats_v2f __attribute__((ext_vector_type(2)));
typedef float hats_v8f __attribute__((ext_vector_type(8)));

// ---------------- kernel: pairwise time-surface accumulation ----------------
// One block handles (batch b, i-slice s). The full per-batch event stream
// (32 KB raw + 8 KB packed codes) is staged into LDS via the CDNA5 async
// global->LDS engine; each event is re-read ~hundreds of times by the
// backward window scans, so LDS residency is the win (320 KB/WGP).
// If pstride != 0, slice s accumulates into its own partial histogram copy
// (contention-free across slices); otherwise directly into the output.
__global__ void __launch_bounds__(256)
hats_pairs(const float* __restrict__ events, const int* __restrict__ lengths,
           float* __restrict__ dest, int T, size_t pstride) {
  __shared__ float    raw[TMAX * 4];   // (x,y,t,p) per event, 32 KB
  __shared__ uint32_t code[TMAX];      // x | y<<8 | (cell*2+p)<<16, 8 KB

  const int b = blockIdx.x / SLICES;
  const int s = blockIdx.x - b * SLICES;
  const float* ev = events + (size_t)b * T * 4;

  __builtin_prefetch(ev + (size_t)threadIdx.x * 4, 0, 1);  // global_prefetch_b8

  // ---- stage raw events into LDS (async DMA path on gfx1250) ----
  for (int e = threadIdx.x; e < T; e += (int)blockDim.x) {
#if HATS_ASYNC
    hats_async_ld_b128(ev + (size_t)e * 4, &raw[e * 4]);
#else
    const float4 v = ((const float4*)ev)[e];
    *(float4*)(&raw[e * 4]) = v;
#endif
  }
#if HATS_ASYNC
  hats_wait_async0();
#endif
  __syncthreads();

  // ---- pack per-event (x, y, cell*2+p) code in LDS ----
  for (int e = threadIdx.x; e < T; e += (int)blockDim.x) {
    const int x = (int)raw[e * 4 + 0];
    const int y = (int)raw[e * 4 + 1];
    const int p = (int)raw[e * 4 + 3];
    const int cell = (y / CELL_K) * GRID_W + (x / CELL_K);
    code[e] = (uint32_t)x | ((uint32_t)y << 8) |
              ((uint32_t)((cell << 1) | p) << 16);
  }
  __syncthreads();

  const int L = min(lengths[b], T);
  const int chunk = (T + SLICES - 1) / SLICES;
  const int i0 = s * chunk;
  const int i1 = min(L, i0 + chunk);
  float* slice_dest = dest + (pstride ? (size_t)s * pstride : (size_t)0);

  for (int i = i0 + (int)threadIdx.x; i < i1; i += (int)blockDim.x) {
    const float    ti  = raw[i * 4 + 2];
    const uint32_t ci  = code[i];
    const int      xi  = (int)(ci & 255u);
    const int      yi  = (int)((ci >> 8) & 255u);
    const uint32_t cpi = ci >> 16;  // cell*2 + p of center event
    float* base = slice_dest + ((size_t)b * (2 * N_CELLS) + cpi) * (SURF * SURF);

    // timestamps are sorted: walk backwards, early-exit past the window.
    // j <= i reproduces the reference's index-causality exactly (incl. ties).
    for (int j = i; j >= 0; --j) {
      const float dt = ti - raw[j * 4 + 2];
      if (dt > DELTA_TW) break;
      const uint32_t cj = code[j];
      if ((cj >> 16) == cpi) {
        const int dx = (int)(cj & 255u) - xi + RADIUS;        // x_j - x_i + R
        const int dy = (int)((cj >> 8) & 255u) - yi + RADIUS; // y_j - y_i + R
        if ((unsigned)dx < SURF && (unsigned)dy < SURF) {
          atomicAdd(base + dy * SURF + dx, __expf(-dt * TAU_INV));
        }
      }
    }
  }
}

// ---------------- kernel: WMMA tree-reduction of partial histograms ----------
// total[n] = sum_k partial[k][n] expressed as ones(16x4) x B(4x16) with
// V_WMMA_F32_16X16X4_F32 (f32 in/out, exact f32 accumulation). Robust to the
// A layout (all-ones) and to any permutation of which k each (VGPR, lane-half)
// slot holds, since we sum over all k; only column = lane%16 striping (matching
// the documented C/D layout) is assumed. Each wave reduces 16 bins; 42336 bins
// per batch = 2646 exact tiles, so EXEC stays all-ones around the WMMAs.
__global__ void __launch_bounds__(256)
hats_reduce(const float* __restrict__ part, const int* __restrict__ counts,
            float* __restrict__ out, int ntot, size_t pstride) {
  const int wave = (int)(blockIdx.x * blockDim.x + threadIdx.x) >> 5;
  const int lane = (int)threadIdx.x & 31;
  const int base = wave * 16;
  if (base >= ntot) return;                 // uniform per-wave
  const int col  = lane & 15;
  const int half = lane >> 4;
  const int g    = base + col;

#if HATS_WMMA
  hats_v8f acc = {};
  hats_v2f a;  a.x = 1.0f; a.y = 1.0f;      // A = ones (layout-invariant)
  hats_v2f b0, b1;
  b0.x = part[(size_t)(0 + half) * pstride + g];   // slots -> partials 0..3
  b0.y = part[(size_t)(2 + half) * pstride + g];
  acc = __builtin_amdgcn_wmma_f32_16x16x4_f32(false, a, false, b0,
                                              (short)0, acc, false, false);
  b1.x = part[(size_t)(4 + half) * pstride + g];   // slots -> partials 4..7
  b1.y = part[(size_t)(6 + half) * pstride + g];
  acc = __builtin_amdgcn_wmma_f32_16x16x4_f32(false, a, false, b1,
                                              (short)0, acc, false, false);
  const float total = acc[0];               // every D row holds the column sum
#else
  float total = 0.0f;
  for (int p = 0; p < NPART; ++p) total += part[(size_t)p * pstride + g];
#endif

  const int c = counts[g / (2 * SURF * SURF)];     // g/98 = b*N_CELLS + cell
  const float v = total * (1.0f / (float)(c > 0 ? c : 1));
  if (half == 0) out[g] = v;
}

// ---------------- kernel: zero accumulators + counts ----------------
__global__ void hats_zero(float* __restrict__ acc, int nacc,
                          int* __restrict__ counts, int ncnt) {
  const int gid = blockIdx.x * blockDim.x + threadIdx.x;
  if (gid < nacc) acc[gid] = 0.0f;
  const int g2 = gid - nacc;
  if (g2 >= 0 && g2 < ncnt) counts[g2] = 0;
}

// ---------------- kernel: per-(b,cell) valid-event counts ----------------
__global__ void hats_count(const float* __restrict__ events,
                           const int* __restrict__ lengths,
                           int* __restrict__ counts, int T, int BT) {
  const int gid = blockIdx.x * blockDim.x + threadIdx.x;
  if (gid >= BT) return;
  const int b = gid / T;
  const int i = gid - b * T;
  if (i < lengths[b]) {
    const int x = (int)events[(size_t)gid * 4 + 0];
    const int y = (int)events[(size_t)gid * 4 + 1];
    const int cell = (y / CELL_K) * GRID_W + (x / CELL_K);
    atomicAdd(&counts[b * N_CELLS + cell], 1);
  }
}

// ---------------- kernel: in-place normalization (fallback path) -------------
__global__ void hats_norm(float* __restrict__ out, const int* __restrict__ counts,
                          int n) {
  const int gid = blockIdx.x * blockDim.x + threadIdx.x;
  if (gid >= n) return;
  const int c = counts[gid / (2 * SURF * SURF)];
  out[gid] *= 1.0f / (float)(c > 0 ? c : 1);
}

// ---------------- host launcher ----------------
extern "C" void kernel_launch(void* const* d_in, const int* in_sizes, int n_in,
                              void* d_out, int out_size, void* d_ws, size_t ws_size,
                              hipStream_t stream) {
  const float* events  = (const float*)d_in[0];  // [B, T, 4] f32
  const int*   lengths = (const int*)d_in[1];    // [B] i32
  float*       out     = (float*)d_out;          // [B, NC, 2, 7, 7] f32

  const int B    = in_sizes[1];
  const int T    = in_sizes[0] / (4 * B);
  const int BT   = B * T;
  const int ncnt = B * N_CELLS;
  const size_t ntot = (size_t)out_size;          // B * NC * 2 * 49
  const size_t need = (NPART * ntot + (size_t)ncnt) * sizeof(float);

  if (ws_size >= need && (ntot % 16) == 0) {
    // ---- partial-copy path: contention-free atomics + WMMA reduction ----
    float* part   = (float*)d_ws;                      // [NPART][ntot]
    int*   counts = (int*)(part + NPART * ntot);       // [B*NC]
    const int nacc = (int)(NPART * ntot);
    hats_zero<<<(nacc + ncnt + 255) / 256, 256, 0, stream>>>(part, nacc,
                                                             counts, ncnt);
    hats_count<<<(BT + 255) / 256, 256, 0, stream>>>(events, lengths, counts,
                                                     T, BT);
    hats_pairs<<<B * SLICES, 256, 0, stream>>>(events, lengths, part, T, ntot);
    const int nthreads = (int)(ntot / 16) * 32;        // one wave per 16 bins
    hats_reduce<<<(nthreads + 255) / 256, 256, 0, stream>>>(part, counts, out,
                                                            (int)ntot, ntot);
  } else {
    // ---- fallback: accumulate straight into out, then normalize ----
    int* counts = (int*)d_ws;
    hats_zero<<<(out_size + ncnt + 255) / 256, 256, 0, stream>>>(out, out_size,
                                                                 counts, ncnt);
    hats_count<<<(BT + 255) / 256, 256, 0, stream>>>(events, lengths, counts,
                                                     T, BT);
    hats_pairs<<<B * SLICES, 256, 0, stream>>>(events, lengths, out, T,
                                               (size_t)0);
    hats_norm<<<(out_size + 255) / 256, 256, 0, stream>>>(out, counts, out_size);
  }
}